// GeometricFeatureExtraction_1168231104867
// MI455X (gfx1250) — compile-verified
//
#include <hip/hip_runtime.h>
#include <math.h>

#define B_   4
#define N_   8192
#define C_   128
#define KNN  20
#define CHUNK 2048

typedef __attribute__((ext_vector_type(2))) float v2f;
typedef __attribute__((ext_vector_type(8))) float v8f;

// Consume one 16-candidate tile's dot products: d' = |p|^2 - 2 p.q, then
// guarded sorted-insert into the per-lane top-20 (all indices static).
#define KNN_CONSUME(DOTV, TIDX)                                              \
  {                                                                          \
    const int gbase_ = ch * CHUNK + (TIDX) * 16 + 8 * half;                  \
    _Pragma("unroll")                                                        \
    for (int r = 0; r < 8; ++r) {                                            \
      const float sq_ = pts[(TIDX) * 16 + 8 * half + r].w; /* LDS bcast */   \
      const float d_ = fmaf(-2.0f, (DOTV)[r], sq_);                          \
      if (d_ < bd[KNN - 1]) {                                                \
        bd[KNN - 1] = d_; bi[KNN - 1] = gbase_ + r;                          \
        _Pragma("unroll")                                                    \
        for (int j = KNN - 1; j > 0; --j) {                                  \
          if (bd[j] < bd[j - 1]) {                                           \
            float td_ = bd[j]; bd[j] = bd[j - 1]; bd[j - 1] = td_;           \
            int   ti_ = bi[j]; bi[j] = bi[j - 1]; bi[j - 1] = ti_;           \
          }                                                                  \
        }                                                                    \
      }                                                                      \
    }                                                                        \
  }

// ---------------------------------------------------------------------------
// Kernel 1: kNN (K=20) via WMMA distance tiles + covariance + smallest
// eigenvector per point. Wave = 16 queries (B operand); candidates streamed
// 16/tile as A operand. Software-pipelined: WMMA for tile t+1 is issued
// before tile t's results are consumed, hiding the XDL->VALU RAW latency.
// ---------------------------------------------------------------------------
__global__ __launch_bounds__(256) void knn_normals_wmma_kernel(
    const float* __restrict__ xyz, float* __restrict__ normals) {
  __shared__ float4 pts[CHUNK];                 // 32 KB, w = |p|^2
  const int tid  = threadIdx.x;
  const int lane = tid & 31;
  const int wave = tid >> 5;
  const int half = lane >> 4;                   // 0: lanes 0-15, 1: lanes 16-31
  const int l16  = lane & 15;
  const int b     = blockIdx.x >> 6;            // 64 blocks of 128 queries per batch
  const int qbase = (blockIdx.x & 63) * 128 + wave * 16;
  const float* xb = xyz + (size_t)b * N_ * 3;

  // B operand (4x16, queries): VGPR0 rows K0(x)/K1(y), VGPR1 rows K2(z)/K3(0)
  const int q = qbase + l16;
  const float qx = xb[3 * q + 0], qy = xb[3 * q + 1], qz = xb[3 * q + 2];
  v2f bq;
  bq.x = half ? qy : qx;
  bq.y = half ? 0.0f : qz;

  float bd[KNN];
  int   bi[KNN];
#pragma unroll
  for (int k = 0; k < KNN; ++k) { bd[k] = 3.0e38f; bi[k] = 0; }

  const v8f z8 = {};
  for (int ch = 0; ch < N_ / CHUNK; ++ch) {
    __syncthreads();
    for (int i = tid; i < CHUNK; i += 256) {
      const int g = ch * CHUNK + i;
      const float px = xb[3 * g + 0], py = xb[3 * g + 1], pz = xb[3 * g + 2];
      pts[i] = make_float4(px, py, pz, px * px + py * py + pz * pz);
    }
    __syncthreads();

    // Prologue: WMMA for tile 0
    // A operand (16x4): lanes 0-15 hold (K0,K1)=(x,y); lanes 16-31 (K2,K3)=(z,0)
    v8f dot_cur;
    {
      const float4 cp = pts[l16];
      v2f a;
      a.x = half ? cp.z : cp.x;
      a.y = half ? 0.0f : cp.y;
      dot_cur = __builtin_amdgcn_wmma_f32_16x16x4_f32(false, a, false, bq,
                                                      (short)0, z8, false, false);
    }
    for (int t = 0; t < CHUNK / 16 - 1; ++t) {
      const float4 cp = pts[(t + 1) * 16 + l16];
      v2f a;
      a.x = half ? cp.z : cp.x;
      a.y = half ? 0.0f : cp.y;
      const v8f dot_next = __builtin_amdgcn_wmma_f32_16x16x4_f32(
          false, a, false, bq, (short)0, z8, false, false);
      KNN_CONSUME(dot_cur, t);      // overlaps with dot_next's WMMA in flight
      dot_cur = dot_next;
    }
    KNN_CONSUME(dot_cur, (CHUNK / 16 - 1));
  }

  // Merge: lo lane absorbs hi lane's sorted 20-list (static shuffles only)
#pragma unroll
  for (int k = 0; k < KNN; ++k) {
    const float pd = __shfl(bd[k], l16 + 16, 32);
    const int   pi = __shfl(bi[k], l16 + 16, 32);
    if (!half && pd < bd[KNN - 1]) {
      bd[KNN - 1] = pd; bi[KNN - 1] = pi;
#pragma unroll
      for (int j = KNN - 1; j > 0; --j) {
        if (bd[j] < bd[j - 1]) {
          float td = bd[j]; bd[j] = bd[j - 1]; bd[j - 1] = td;
          int   ti = bi[j]; bi[j] = bi[j - 1]; bi[j - 1] = ti;
        }
      }
    }
  }
  if (half) return;   // epilogue on lo lanes only (one query each)

  // 3x3 covariance of neighbors centered on query point (xyz is L2-resident)
  float cxx = 0.f, cxy = 0.f, cxz = 0.f, cyy = 0.f, cyz = 0.f, czz = 0.f;
#pragma unroll
  for (int k = 0; k < KNN; ++k) {
    const int idx = bi[k];
    const float dx = xb[3 * idx + 0] - qx;
    const float dy = xb[3 * idx + 1] - qy;
    const float dz = xb[3 * idx + 2] - qz;
    cxx += dx * dx; cxy += dx * dy; cxz += dx * dz;
    cyy += dy * dy; cyz += dy * dz; czz += dz * dz;
  }

  // Smallest eigenvalue of symmetric 3x3 via trig (Cardano) formula
  const float p1 = cxy * cxy + cxz * cxz + cyz * cyz;
  const float qm = (cxx + cyy + czz) * (1.0f / 3.0f);
  const float b00 = cxx - qm, b11 = cyy - qm, b22 = czz - qm;
  const float p2 = b00 * b00 + b11 * b11 + b22 * b22 + 2.0f * p1;
  const float pp = sqrtf(p2 * (1.0f / 6.0f));
  float eig = qm;
  if (pp > 1e-20f) {
    const float inv = 1.0f / pp;
    const float c00 = b00 * inv, c01 = cxy * inv, c02 = cxz * inv;
    const float c11 = b11 * inv, c12 = cyz * inv, c22 = b22 * inv;
    float detB = c00 * (c11 * c22 - c12 * c12)
               - c01 * (c01 * c22 - c12 * c02)
               + c02 * (c01 * c12 - c11 * c02);
    float r = fminf(1.0f, fmaxf(-1.0f, 0.5f * detB));
    const float phi = acosf(r) * (1.0f / 3.0f);
    eig = qm + 2.0f * pp * cosf(phi + 2.0943951023931953f);  // smallest
  }

  // Eigenvector: largest cross product of rows of (A - eig*I)
  const float m00 = cxx - eig, m11 = cyy - eig, m22 = czz - eig;
  float v0x = cxy * cyz - cxz * m11, v0y = cxz * cxy - m00 * cyz, v0z = m00 * m11 - cxy * cxy;
  float v1x = cxy * m22 - cxz * cyz, v1y = cxz * cxz - m00 * m22, v1z = m00 * cyz - cxy * cxz;
  float v2x = m11 * m22 - cyz * cyz, v2y = cyz * cxz - cxy * m22, v2z = cxy * cyz - m11 * cxz;
  const float n0s = v0x * v0x + v0y * v0y + v0z * v0z;
  const float n1s = v1x * v1x + v1y * v1y + v1z * v1z;
  const float n2s = v2x * v2x + v2y * v2y + v2z * v2z;
  float vx = v0x, vy = v0y, vz = v0z, ns = n0s;
  if (n1s > ns) { vx = v1x; vy = v1y; vz = v1z; ns = n1s; }
  if (n2s > ns) { vx = v2x; vy = v2y; vz = v2z; ns = n2s; }
  if (ns < 1e-30f) { vx = 0.0f; vy = 0.0f; vz = 1.0f; ns = 1.0f; }
  const float rn = rsqrtf(ns);
  normals[((size_t)b * 3 + 0) * N_ + q] = vx * rn;
  normals[((size_t)b * 3 + 1) * N_ + q] = vy * rn;
  normals[((size_t)b * 3 + 2) * N_ + q] = vz * rn;
}

// ---------------------------------------------------------------------------
// Kernel 2: h = w1[:, :128] @ x  (+ rank-3 normals term + bias), f32 WMMA.
// ---------------------------------------------------------------------------
__global__ __launch_bounds__(256) void gemm1_wmma_kernel(
    const float* __restrict__ x, const float* __restrict__ normals,
    const float* __restrict__ w1, const float* __restrict__ b1,
    float* __restrict__ h) {
  const int lane = threadIdx.x & 31;
  const int wave = threadIdx.x >> 5;
  const int half = lane >> 4;
  const int l16  = lane & 15;
  const int n0   = (blockIdx.x & 511) * 16;
  const int b    = blockIdx.x >> 9;
  const int o0   = wave * 16;

  v2f areg[32];
  {
    const float* wr = w1 + (size_t)(o0 + l16) * 131;
#pragma unroll
    for (int s = 0; s < 32; ++s) {
      const int k = s * 4 + half * 2;
      areg[s].x = wr[k];
      areg[s].y = wr[k + 1];
    }
  }

  v8f acc = {};
  const float* xb = x + (size_t)b * C_ * N_;
#pragma unroll
  for (int s = 0; s < 32; ++s) {
    const int kk = s * 4;
    if (kk + 8 < 128)
      __builtin_prefetch(&xb[(size_t)(kk + 8 + half) * N_ + n0 + l16], 0, 3);
    v2f bb;
    bb.x = xb[(size_t)(kk + half) * N_ + n0 + l16];
    bb.y = xb[(size_t)(kk + 2 + half) * N_ + n0 + l16];
    acc = __builtin_amdgcn_wmma_f32_16x16x4_f32(false, areg[s], false, bb,
                                                (short)0, acc, false, false);
  }

  const int n = n0 + l16;
  const float nx = normals[((size_t)b * 3 + 0) * N_ + n];
  const float ny = normals[((size_t)b * 3 + 1) * N_ + n];
  const float nz = normals[((size_t)b * 3 + 2) * N_ + n];
  float* hb = h + (size_t)b * C_ * N_;
#pragma unroll
  for (int r = 0; r < 8; ++r) {
    const int orow = o0 + half * 8 + r;
    const float* wr = w1 + (size_t)orow * 131;
    const float v = acc[r] + wr[128] * nx + wr[129] * ny + wr[130] * nz + b1[orow];
    hb[(size_t)orow * N_ + n] = v;
  }
}

// ---------------------------------------------------------------------------
// Kernel 3: per-channel BN statistics -> scale/shift (training-mode, biased var)
// ---------------------------------------------------------------------------
__global__ __launch_bounds__(256) void bn_stats_kernel(
    const float* __restrict__ h, const float* __restrict__ gamma,
    const float* __restrict__ beta, float* __restrict__ sA, float* __restrict__ sB) {
  const int c = blockIdx.x;
  const int tid = threadIdx.x;
  float s = 0.0f, ss = 0.0f;
  for (int i = tid; i < B_ * N_; i += 256) {
    const int b = i >> 13;
    const int n = i & (N_ - 1);
    const float v = h[((size_t)b * C_ + c) * N_ + n];
    s += v; ss += v * v;
  }
  __shared__ float rs[256], rq[256];
  rs[tid] = s; rq[tid] = ss;
  __syncthreads();
  for (int off = 128; off > 0; off >>= 1) {
    if (tid < off) { rs[tid] += rs[tid + off]; rq[tid] += rq[tid + off]; }
    __syncthreads();
  }
  if (tid == 0) {
    const float inv_n = 1.0f / (float)(B_ * N_);
    const float mean = rs[0] * inv_n;
    const float var  = rq[0] * inv_n - mean * mean;
    const float inv  = rsqrtf(var + 1e-5f);
    const float g = gamma[c];
    sA[c] = g * inv;
    sB[c] = beta[c] - g * inv * mean;
  }
}

// ---------------------------------------------------------------------------
// Kernel 4: out = w2 @ relu(scale*h + shift) + b2, f32 WMMA, BN+ReLU fused
// into the B-operand load.
// ---------------------------------------------------------------------------
__global__ __launch_bounds__(256) void gemm2_wmma_kernel(
    const float* __restrict__ h, const float* __restrict__ sA,
    const float* __restrict__ sB, const float* __restrict__ w2,
    const float* __restrict__ b2, float* __restrict__ out) {
  const int lane = threadIdx.x & 31;
  const int wave = threadIdx.x >> 5;
  const int half = lane >> 4;
  const int l16  = lane & 15;
  const int n0   = (blockIdx.x & 511) * 16;
  const int b    = blockIdx.x >> 9;
  const int o0   = wave * 16;

  v2f areg[32];
  {
    const float* wr = w2 + (size_t)(o0 + l16) * C_;
#pragma unroll
    for (int s = 0; s < 32; ++s) {
      const int k = s * 4 + half * 2;
      areg[s].x = wr[k];
      areg[s].y = wr[k + 1];
    }
  }

  v8f acc = {};
  const float* hb = h + (size_t)b * C_ * N_;
  const int n = n0 + l16;
#pragma unroll
  for (int s = 0; s < 32; ++s) {
    const int kk = s * 4;
    const int k0 = kk + half;
    const int k1 = kk + 2 + half;
    const float a0 = fmaxf(sA[k0] * hb[(size_t)k0 * N_ + n] + sB[k0], 0.0f);
    const float a1 = fmaxf(sA[k1] * hb[(size_t)k1 * N_ + n] + sB[k1], 0.0f);
    v2f bb; bb.x = a0; bb.y = a1;
    acc = __builtin_amdgcn_wmma_f32_16x16x4_f32(false, areg[s], false, bb,
                                                (short)0, acc, false, false);
  }

  float* ob = out + (size_t)b * C_ * N_;
#pragma unroll
  for (int r = 0; r < 8; ++r) {
    const int orow = o0 + half * 8 + r;
    ob[(size_t)orow * N_ + n] = acc[r] + b2[orow];
  }
}

// ---------------------------------------------------------------------------
extern "C" void kernel_launch(void* const* d_in, const int* in_sizes, int n_in,
                              void* d_out, int out_size, void* d_ws, size_t ws_size,
                              hipStream_t stream) {
  const float* x     = (const float*)d_in[0];   // [4,128,8192]
  const float* xyz   = (const float*)d_in[1];   // [4,8192,3]
  const float* w1    = (const float*)d_in[2];   // [128,131]
  const float* b1    = (const float*)d_in[3];   // [128]
  const float* gamma = (const float*)d_in[4];   // [128]
  const float* beta  = (const float*)d_in[5];   // [128]
  const float* w2    = (const float*)d_in[6];   // [128,128]
  const float* b2    = (const float*)d_in[7];   // [128]
  float* out = (float*)d_out;                   // [4,128,8192]

  float* normals = (float*)d_ws;                        // 4*3*8192 floats
  float* hbuf    = normals + (size_t)B_ * 3 * N_;       // 4*128*8192 floats
  float* scaleA  = hbuf + (size_t)B_ * C_ * N_;         // 128
  float* shiftB  = scaleA + C_;                         // 128

  knn_normals_wmma_kernel<<<B_ * (N_ / 128), 256, 0, stream>>>(xyz, normals);
  gemm1_wmma_kernel<<<B_ * (N_ / 16), 256, 0, stream>>>(x, normals, w1, b1, hbuf);
  bn_stats_kernel<<<C_, 256, 0, stream>>>(hbuf, gamma, beta, scaleA, shiftB);
  gemm2_wmma_kernel<<<B_ * (N_ / 16), 256, 0, stream>>>(hbuf, scaleA, shiftB, w2, b2, out);
}